// PoissonModel_9586367004740
// MI455X (gfx1250) — compile-verified
//
#include <hip/hip_runtime.h>
#include <math.h>

// ---------------------------------------------------------------------------
// FEM Poisson solve (Q1, uniform 96x96 grid, variable kappa) + CG in LDS,
// then bilinear interpolation at NPTS query points.
//
// MI455X (gfx1250) design:
//  * Whole CG state (u, r, p, Ap, kappa) in one WGP's 320KB LDS; one
//    workgroup of 1024 threads (32 wave32s) runs all CG iterations.
//  * y = A*p is per-element D(16x16) = A(16x4) x B(4x16) with
//    V_WMMA_F32_16X16X4_F32 (A = 16 same-color elements' kappa*p gathers,
//    B = constant local stiffness Kloc, zero-padded).
//  * 4-color element schedule -> scatter has no address collisions; scatter
//    is 8 ds_add_f32 with immediate offsets off one base address (no-return
//    LDS float atomics; out-of-range lanes contribute exact +/-0 into a
//    padded region -> branch- and clamp-free).
//  * CG vector ops: 3x3 node tile per thread -> all addresses are
//    base + compile-time constant (DS immediate offsets), no div/mod,
//    interior masks hoisted out of the iteration loop.
//  * Interpolation: separate full-device memory-bound kernel.
// ---------------------------------------------------------------------------

typedef float v2f __attribute__((ext_vector_type(2)));
typedef float v8f __attribute__((ext_vector_type(8)));

#define NG 96            // nodes per dimension (N+1)
#define NE 95            // elements per dimension (N)
#define NN (NG * NG)     // 9216 nodes
#define NELEM (NE * NE)  // 9025 elements
#define NTH 1024
#define NWAVES (NTH / 32)
#define MAXIT 5000

// Two-barrier block-wide allreduce with a dedicated result slot; safe to call
// back-to-back (no wave can re-enter the next reduction's write phase before
// every wave has passed this one's final barrier).
__device__ __forceinline__ float block_allreduce(float v, float* wred, float* res,
                                                 int lane, int wave) {
#pragma unroll
    for (int o = 16; o > 0; o >>= 1) v += __shfl_xor(v, o, 32);
    if (lane == 0) wred[wave] = v;
    __syncthreads();
    if (wave == 0) {
        float s = wred[lane];
#pragma unroll
        for (int o = 16; o > 0; o >>= 1) s += __shfl_xor(s, o, 32);
        if (lane == 0) *res = s;
    }
    __syncthreads();
    return *res;
}

__global__ void __launch_bounds__(NTH, 1)
fem_cg_solver(const float* __restrict__ params, float* __restrict__ u_out) {
    __shared__ float s_u[NN];
    __shared__ float s_r[NN];
    __shared__ float s_p[NN];
    __shared__ float s_y[NN + 64];   // +64: sink for the single +/-0 overflow add
    __shared__ float s_kap[NELEM];
    __shared__ float s_sx[NE];
    __shared__ float s_red[NWAVES];
    __shared__ float s_res;

    const int tid  = threadIdx.x;
    const int lane = tid & 31;
    const int wave = tid >> 5;

    const float A0 = -3.0f;
    const float H  = 6.0f / 95.0f;
    const float PI = 3.14159265358979323846f;

    const float q0 = params[0], q1 = params[1], q2 = params[2], q3 = params[3];

    // --- per-cell kappa (midpoint quadrature) and separable forcing factors ---
    for (int e = tid; e < NELEM; e += NTH) {
        int ei = e / NE, ej = e % NE;
        float cx = A0 + H * ((float)ei + 0.5f);
        float cy = A0 + H * ((float)ej + 0.5f);
        float dx = cx - q2, dy = cy - q3;
        s_kap[e] = q0 + q1 * expf(-(dx * dx + dy * dy));
    }
    for (int k = tid; k < NE; k += NTH) {
        s_sx[k] = sinf(PI * (A0 + H * ((float)k + 0.5f)));
    }
    __syncthreads();

    // --- 3x3 node tile per thread: nodes (3*ti+di, 3*tj+dj) -----------------
    const int ti = tid >> 5;          // 0..31
    const int tj = tid & 31;          // 0..31
    const int i0 = 3 * ti, j0 = 3 * tj;
    const int nb = i0 * NG + j0;      // base node index
    bool rok[3], cok[3];
#pragma unroll
    for (int d = 0; d < 3; ++d) {
        rok[d] = (i0 + d >= 1) && (i0 + d <= NG - 2);
        cok[d] = (j0 + d >= 1) && (j0 + d <= NG - 2);
    }

    // --- load vector b (interior only): F_ij = h^2/4*(sx[i-1]+sx[i])(sx[j-1]+sx[j])
    float bb = 0.f;
#pragma unroll
    for (int di = 0; di < 3; ++di) {
#pragma unroll
        for (int dj = 0; dj < 3; ++dj) {
            int i = i0 + di, j = j0 + dj;
            int im = (i > 0) ? i - 1 : 0;   // clamped (value unused when masked)
            int jm = (j > 0) ? j - 1 : 0;
            float bv = 0.25f * H * H * (s_sx[im] + s_sx[i]) * (s_sx[jm] + s_sx[j]);
            if (!(rok[di] && cok[dj])) bv = 0.f;
            int n = nb + di * NG + dj;
            s_u[n] = 0.f;
            s_r[n] = bv;   // x0 = 0 -> r = b
            s_p[n] = bv;
            s_y[n] = 0.f;
            bb += bv * bv;
        }
    }
    float rs = block_allreduce(bb, s_red, &s_res, lane, wave);
    const float atol2 = 1e-10f * rs;  // (tol=1e-5)^2 * (b.b), jax cg criterion

    // --- constant B fragment (4x16) = Kloc zero-padded.
    //     Kloc[k][l] = (k==l ? 4 : ((k^l)==2 ? -2 : -1)) / 6
    //     f32 B layout: lanes 0-15 col N=lane, VGPR0=K0,VGPR1=K1;
    //                   lanes 16-31 col N=lane-16, VGPR0=K2,VGPR1=K3.
    const int ncol = lane & 15;
    const int kb   = (lane >> 4) << 1;  // 0 or 2
    v2f bmat;
    {
        const int k0 = kb, k1 = kb + 1;
        bmat.x = (ncol < 4) ? ((k0 == ncol) ? (4.f / 6.f)
                                            : (((k0 ^ ncol) == 2) ? (-2.f / 6.f) : (-1.f / 6.f)))
                            : 0.f;
        bmat.y = (ncol < 4) ? ((k1 == ncol) ? (4.f / 6.f)
                                            : (((k1 ^ ncol) == 2) ? (-2.f / 6.f) : (-1.f / 6.f)))
                            : 0.f;
    }

    // local dof k node offsets: 0:(0,0) 1:(+1,0) 2:(+1,+1) 3:(0,+1)
    const int oA0 = (kb == 0) ? 0 : (NG + 1);  // this lane's K, K+1 offsets (A side)
    const int oA1 = (kb == 0) ? NG : 1;

    // scatter constants (lanes with nn = lane&15 < 4 do the scatter)
    const int nn_s = lane & 15;
    const int mb_s = (lane >> 4) << 3;                 // 0 or 8
    const int di_s = ((nn_s + 1) >> 1) & 1;            // row offset of dof nn
    const int dj_s = nn_s >> 1;                        // col offset of dof nn
    const bool do_scatter = (nn_s < 4);

    int it = 0;
    while (rs > atol2 && it < MAXIT) {
        // ---------------- y = A * p (WMMA per 16-element tile) ----------------
        // (s_y was zeroed at end of previous iteration / at init)
#pragma unroll
        for (int color = 0; color < 4; ++color) {
            const int cxo = color >> 1, cyo = color & 1;     // ei, ej parity
            const int nA  = 48 - cxo;                        // # element rows
            const int nBc = 48 - cyo;                        // # elements per row
            const int ntiles = nA * 3;                       // 3 chunks of 16 per row

            for (int tile = wave; tile < ntiles; tile += NWAVES) {
                const int a = tile / 3;          // element row (const division)
                const int c = tile - 3 * a;      // chunk in row
                const int ei = cxo + 2 * a;

                // A row m = lane&15 -> element eb = c*16+m in this row
                const int m  = lane & 15;
                int eb  = c * 16 + m;
                int ebc = (eb < nBc) ? eb : (nBc - 1);       // safe load index
                int ej  = cyo + 2 * ebc;
                float kv = s_kap[ei * NE + ej];              // unconditional load
                kv = (eb < nBc) ? kv : 0.f;                  // select, no branch
                int base = ei * NG + ej;

                v2f av;
                av.x = kv * s_p[base + oA0];
                av.y = kv * s_p[base + oA1];

                v8f acc = {};
                // D = A(16x4) x B(4x16) + 0 ; EXEC full (uniform loop)
                acc = __builtin_amdgcn_wmma_f32_16x16x4_f32(
                    false, av, false, bmat, (short)0, acc, false, false);

                // D layout: lanes 0-15: VGPR v -> (M=v, N=lane);
                //           lanes 16-31: VGPR v -> (M=v+8, N=lane-16).
                // Columns N=0..3 hold the 4 nodal contributions of element M.
                // Same-color elements share no nodes -> collision-free adds.
                // Addresses form base + 8*v bytes -> DS immediate offsets.
                // The lone out-of-range lane adds exact +/-0 (kv=0 row) into
                // the next row / the 64-float pad: numerically a no-op.
                if (do_scatter) {
                    float* py = &s_y[(ei + di_s) * NG + cyo + dj_s + 2 * (c * 16 + mb_s)];
#pragma unroll
                    for (int v = 0; v < 8; ++v) {
                        atomicAdd(py + 2 * v, acc[v]);
                    }
                }
            }
            __syncthreads();
        }

        // ---------------- CG scalar updates -----------------------------------
        // p == 0 on the boundary, so p.y needs no interior mask.
        float pap = 0.f;
#pragma unroll
        for (int di = 0; di < 3; ++di)
#pragma unroll
            for (int dj = 0; dj < 3; ++dj) {
                int n = nb + di * NG + dj;
                pap += s_p[n] * s_y[n];
            }
        pap = block_allreduce(pap, s_red, &s_res, lane, wave);
        float alpha = rs / pap;

        float rsn = 0.f;
#pragma unroll
        for (int di = 0; di < 3; ++di)
#pragma unroll
            for (int dj = 0; dj < 3; ++dj) {
                int n = nb + di * NG + dj;
                float yv = (rok[di] && cok[dj]) ? s_y[n] : 0.f;  // mask boundary y
                s_u[n] += alpha * s_p[n];                        // p==0 on boundary
                float rn = s_r[n] - alpha * yv;                  // r stays 0 there
                s_r[n] = rn;
                rsn += rn * rn;
            }
        rsn = block_allreduce(rsn, s_red, &s_res, lane, wave);
        float beta = rsn / rs;

        // p update (r==0, p==0 on boundary -> stays 0); zero y for next apply
#pragma unroll
        for (int di = 0; di < 3; ++di)
#pragma unroll
            for (int dj = 0; dj < 3; ++dj) {
                int n = nb + di * NG + dj;
                s_p[n] = s_r[n] + beta * s_p[n];
                s_y[n] = 0.f;
            }
        __syncthreads();

        rs = rsn;
        ++it;
    }

#pragma unroll
    for (int di = 0; di < 3; ++di)
#pragma unroll
        for (int dj = 0; dj < 3; ++dj) {
            int n = nb + di * NG + dj;
            u_out[n] = s_u[n];
        }
}

// ---------------------------------------------------------------------------
// Bilinear interpolation at the query points (full-device, memory-bound).
// ---------------------------------------------------------------------------
__global__ void interp_kernel(const float* __restrict__ ug,
                              const float* __restrict__ xs,
                              const float* __restrict__ ys,
                              float* __restrict__ out, int n) {
    int idx = blockIdx.x * blockDim.x + threadIdx.x;
    if (idx >= n) return;
    const float A0 = -3.0f;
    const float INVH = 95.0f / 6.0f;
    float tx = (xs[idx] - A0) * INVH;
    float ty = (ys[idx] - A0) * INVH;
    float fxf = fminf(fmaxf(floorf(tx), 0.f), 94.f);
    float fyf = fminf(fmaxf(floorf(ty), 0.f), 94.f);
    int ix = (int)fxf, iy = (int)fyf;
    float fx = tx - fxf, fy = ty - fyf;
    int row = ix * NG + iy;
    float v00 = ug[row];
    float v10 = ug[row + NG];
    float v01 = ug[row + 1];
    float v11 = ug[row + NG + 1];
    out[idx] = v00 * (1.f - fx) * (1.f - fy) + v10 * fx * (1.f - fy)
             + v01 * (1.f - fx) * fy + v11 * fx * fy;
}

extern "C" void kernel_launch(void* const* d_in, const int* in_sizes, int n_in,
                              void* d_out, int out_size, void* d_ws, size_t ws_size,
                              hipStream_t stream) {
    const float* params = (const float*)d_in[0];
    const float* xs     = (const float*)d_in[1];
    const float* ys     = (const float*)d_in[2];
    float* out = (float*)d_out;
    float* ug  = (float*)d_ws;  // 9216 floats: solved grid values

    fem_cg_solver<<<1, NTH, 0, stream>>>(params, ug);

    int npts = in_sizes[1];
    interp_kernel<<<(npts + 255) / 256, 256, 0, stream>>>(ug, xs, ys, out, npts);
}